// GAT_84859963834575
// MI455X (gfx1250) — compile-verified
//
#include <hip/hip_runtime.h>
#include <hip/hip_bf16.h>

typedef float v2f __attribute__((ext_vector_type(2)));
typedef float v8f __attribute__((ext_vector_type(8)));

#define N_NODES 50000
#define E_EDGES 500000
#define E2_EDGES 62500
#define ET_EDGES (E_EDGES + E2_EDGES)
#define NREL 1000
#define ALPHA 0.2f
#define KDIM 128

// ---------------------------------------------------------------------------
// f32 WMMA GEMM: C[m][coff+n] = sum_k A[m*lda+k] * B[n*ldbn + k*LDBK], K=128.
// One wave computes a 16 x (NT*16) tile; A fragment reused across NT WMMAs.
// Software-pipelined one k-step ahead so loads overlap the WMMA chain.
// OOB M rows: A row pointer clamped (harmless garbage rows, never stored).
// ---------------------------------------------------------------------------
template <int LDBK, int NT>
__global__ void gat_wmma_gemm_f32(const float* __restrict__ A,
                                  const float* __restrict__ B,
                                  float* __restrict__ C,
                                  int M, int lda, int ldbn, int ldc, int coff,
                                  int mtiles, int ngroups) {
    int wave = threadIdx.x >> 5;
    long tile = (long)blockIdx.x * (blockDim.x >> 5) + wave;
    long tiles_total = (long)mtiles * ngroups;
    if (tile >= tiles_total) return;                 // uniform per wave
    int ng = (int)(tile / mtiles);                   // adjacent waves share B
    int mt = (int)(tile % mtiles);

    int lane = threadIdx.x & 31;
    int half = lane >> 4;      // 0: K={k0,k0+1}, 1: K={k0+2,k0+3}
    int l16  = lane & 15;

    int arow = mt * 16 + l16;
    int arow_c = arow < M ? arow : (M - 1);          // clamp; OOB rows unused
    const float* Ap = A + (size_t)arow_c * lda + half * 2;
    const float* Bp[NT];
#pragma unroll
    for (int s = 0; s < NT; ++s)
        Bp[s] = B + (size_t)(ng * NT * 16 + s * 16 + l16) * ldbn
                  + (size_t)(half * 2) * LDBK;

    v8f acc[NT];
#pragma unroll
    for (int s = 0; s < NT; ++s) acc[s] = (v8f){0.f,0.f,0.f,0.f,0.f,0.f,0.f,0.f};

    // pipeline prologue: stage-0 fragments
    v2f a_cur = *(const v2f*)(Ap);
    v2f b_cur[NT];
#pragma unroll
    for (int s = 0; s < NT; ++s) {
        if constexpr (LDBK == 1) b_cur[s] = *(const v2f*)(Bp[s]);
        else { b_cur[s][0] = Bp[s][0]; b_cur[s][1] = Bp[s][LDBK]; }
    }

#pragma unroll 4
    for (int k0 = 0; k0 < KDIM; k0 += 4) {
        int kn = (k0 + 4 < KDIM) ? (k0 + 4) : 0;     // wrap avoids OOB prefetch
        v2f a_nxt = *(const v2f*)(Ap + kn);
        v2f b_nxt[NT];
#pragma unroll
        for (int s = 0; s < NT; ++s) {
            if constexpr (LDBK == 1) b_nxt[s] = *(const v2f*)(Bp[s] + kn);
            else {
                b_nxt[s][0] = Bp[s][(size_t)kn * LDBK];
                b_nxt[s][1] = Bp[s][(size_t)kn * LDBK + LDBK];
            }
        }
#pragma unroll
        for (int s = 0; s < NT; ++s)
            acc[s] = __builtin_amdgcn_wmma_f32_16x16x4_f32(false, a_cur, false,
                                                           b_cur[s], (short)0,
                                                           acc[s], false, false);
        a_cur = a_nxt;
#pragma unroll
        for (int s = 0; s < NT; ++s) b_cur[s] = b_nxt[s];
    }

#pragma unroll
    for (int s = 0; s < NT; ++s) {
        int ccol = coff + ng * NT * 16 + s * 16 + l16;
#pragma unroll
        for (int r = 0; r < 8; ++r) {
            int crow = mt * 16 + r + half * 8;
            if (crow < M) C[(size_t)crow * ldc + ccol] = acc[s][r];
        }
    }
}

// ---------------------------------------------------------------------------
// Layer-1 edge kernel: one wave per edge, both heads.
// xnode: N x 256  [h0a | h0b | h1a | h1b], ec: E x 128 [h0c | h1c],
// relc: NREL x 128, w2h: 2 x 64, num1: N x 128, denom1: N x 2
// ---------------------------------------------------------------------------
__global__ void gat_edge1(const int* __restrict__ el, const int* __restrict__ eln,
                          const int* __restrict__ etn,
                          const float* __restrict__ xnode,
                          const float* __restrict__ ec,
                          const float* __restrict__ relc,
                          const float* __restrict__ w2h,
                          float* __restrict__ num1, float* __restrict__ denom1) {
    int e = blockIdx.x * (blockDim.x >> 5) + (threadIdx.x >> 5);
    if (e >= ET_EDGES) return;
    int lane = threadIdx.x & 31;

    int s, d, t0 = 0, t1 = 0;
    bool direct = (e < E_EDGES);
    if (direct) {
        s = el[e]; d = el[E_EDGES + e];
    } else {
        int j = e - E_EDGES;
        s = eln[j]; d = eln[E2_EDGES + j];
        t0 = etn[2 * j]; t1 = etn[2 * j + 1];
    }

    int k = lane * 2;
#pragma unroll
    for (int h = 0; h < 2; ++h) {
        float2 xa = *(const float2*)(xnode + (size_t)s * 256 + h * 128 + k);
        float2 xb = *(const float2*)(xnode + (size_t)d * 256 + h * 128 + 64 + k);
        float e0, e1;
        if (direct) {
            float2 ev = *(const float2*)(ec + (size_t)e * 128 + h * 64 + k);
            e0 = ev.x; e1 = ev.y;
        } else {
            float2 a0 = *(const float2*)(relc + (size_t)t0 * 128 + h * 64 + k);
            float2 a1 = *(const float2*)(relc + (size_t)t1 * 128 + h * 64 + k);
            e0 = a0.x + a1.x; e1 = a0.y + a1.y;
        }
        float m0 = xa.x + xb.x + e0;
        float m1 = xa.y + xb.y + e1;
        float2 wv = *(const float2*)(w2h + h * 64 + k);
        float p = m0 * wv.x + m1 * wv.y;
#pragma unroll
        for (int off = 16; off > 0; off >>= 1) p += __shfl_xor(p, off, 32);
        float lr = p > 0.f ? p : ALPHA * p;
        float eev = __expf(-lr);
        float* dp = num1 + (size_t)s * 128 + h * 64 + k;
        atomicAdd(dp,     eev * m0);
        atomicAdd(dp + 1, eev * m1);
        if (lane == 0) atomicAdd(denom1 + (size_t)s * 2 + h, eev);
    }
}

__global__ void gat_fin1(float* __restrict__ num1, const float* __restrict__ denom1) {
    int i = blockIdx.x * blockDim.x + threadIdx.x;
    if (i >= N_NODES * 128) return;
    int n = i >> 7;
    int h = (i & 127) >> 6;
    float dnm = denom1[n * 2 + h];
    dnm = dnm > 1e-12f ? dnm : 1e-12f;
    num1[i] = num1[i] / dnm;
}

// ---------------------------------------------------------------------------
// Layer-2 edge kernel: one wave per edge, 128-dim, 4 components per lane.
// xab2: N x 256 [a2 | b2], relW2: NREL x 128, w2o: 128
// ---------------------------------------------------------------------------
__global__ void gat_edge2(const int* __restrict__ el, const int* __restrict__ eln,
                          const int* __restrict__ et, const int* __restrict__ etn,
                          const float* __restrict__ xab2,
                          const float* __restrict__ relW2,
                          const float* __restrict__ w2o,
                          float* __restrict__ num2, float* __restrict__ denom2) {
    int e = blockIdx.x * (blockDim.x >> 5) + (threadIdx.x >> 5);
    if (e >= ET_EDGES) return;
    int lane = threadIdx.x & 31;

    int s, d;
    float4 ev;
    int k = lane * 4;
    if (e < E_EDGES) {
        s = el[e]; d = el[E_EDGES + e];
        ev = *(const float4*)(relW2 + (size_t)et[e] * 128 + k);
    } else {
        int j = e - E_EDGES;
        s = eln[j]; d = eln[E2_EDGES + j];
        int t0 = etn[2 * j], t1 = etn[2 * j + 1];
        float4 a0 = *(const float4*)(relW2 + (size_t)t0 * 128 + k);
        float4 a1 = *(const float4*)(relW2 + (size_t)t1 * 128 + k);
        ev.x = a0.x + a1.x; ev.y = a0.y + a1.y;
        ev.z = a0.z + a1.z; ev.w = a0.w + a1.w;
    }
    float4 xa = *(const float4*)(xab2 + (size_t)s * 256 + k);
    float4 xb = *(const float4*)(xab2 + (size_t)d * 256 + 128 + k);
    float m0 = xa.x + xb.x + ev.x;
    float m1 = xa.y + xb.y + ev.y;
    float m2 = xa.z + xb.z + ev.z;
    float m3 = xa.w + xb.w + ev.w;
    float4 wv = *(const float4*)(w2o + k);
    float p = m0 * wv.x + m1 * wv.y + m2 * wv.z + m3 * wv.w;
#pragma unroll
    for (int off = 16; off > 0; off >>= 1) p += __shfl_xor(p, off, 32);
    float lr = p > 0.f ? p : ALPHA * p;
    float eev = __expf(-lr);
    float* dp = num2 + (size_t)s * 128 + k;
    atomicAdd(dp,     eev * m0);
    atomicAdd(dp + 1, eev * m1);
    atomicAdd(dp + 2, eev * m2);
    atomicAdd(dp + 3, eev * m3);
    if (lane == 0) atomicAdd(denom2 + s, eev);
}

__global__ void gat_fin2(const float* __restrict__ num2,
                         const float* __restrict__ denom2,
                         float* __restrict__ out) {
    int i = blockIdx.x * blockDim.x + threadIdx.x;
    if (i >= N_NODES * 128) return;
    int n = i >> 7;
    float dnm = denom2[n];
    dnm = dnm > 1e-12f ? dnm : 1e-12f;
    float h = num2[i] / dnm;
    float y = h > 0.f ? h : (__expf(h) - 1.f);   // elu inside gat_layer (out_layer)
    y = y > 0.f ? y : (__expf(y) - 1.f);         // elu in reference outer
    out[i] = y;
}

// ---------------------------------------------------------------------------
template <int LDBK>
static inline void launch_gemm(hipStream_t st, const float* A, const float* B,
                               float* C, int M, int N,
                               int lda, int ldbn, int ldc, int coff) {
    constexpr int NT = 4;                 // 16 x 64 tile per wave
    int mtiles  = (M + 15) / 16;
    int ngroups = N / (NT * 16);
    long tiles = (long)mtiles * ngroups;
    int blocks = (int)((tiles + 7) / 8);
    gat_wmma_gemm_f32<LDBK, NT><<<blocks, 256, 0, st>>>(A, B, C, M, lda, ldbn,
                                                        ldc, coff, mtiles,
                                                        ngroups);
}

extern "C" void kernel_launch(void* const* d_in, const int* in_sizes, int n_in,
                              void* d_out, int out_size, void* d_ws, size_t ws_size,
                              hipStream_t stream) {
    const float* x    = (const float*)d_in[0];   // N x 128
    const float* rel  = (const float*)d_in[1];   // NREL x 128
    const int*   el   = (const int*)d_in[2];     // 2 x E
    const int*   et   = (const int*)d_in[3];     // E
    const float* eemb = (const float*)d_in[4];   // E x 128
    const int*   eln  = (const int*)d_in[5];     // 2 x E2
    const int*   etn  = (const int*)d_in[6];     // E2 x 2
    const float* w1h  = (const float*)d_in[8];   // 2 x 64 x 384
    const float* w2h  = (const float*)d_in[9];   // 2 x 1 x 64
    const float* w1o  = (const float*)d_in[10];  // 128 x 384
    const float* w2o  = (const float*)d_in[11];  // 1 x 128
    const float* WR   = (const float*)d_in[12];  // 128 x 128
    float* out = (float*)d_out;                  // [N*128 entity | NREL*128 relation]

    // workspace layout (floats)
    float* ws = (float*)d_ws;
    float* xnode  = ws;                             // N x 256
    float* ec     = xnode  + (size_t)N_NODES * 256; // E x 128
    float* relc   = ec     + (size_t)E_EDGES * 128; // NREL x 128
    float* num1   = relc   + (size_t)NREL * 128;    // N x 128 (becomes layer_x)
    float* denom1 = num1   + (size_t)N_NODES * 128; // N x 2
    float* xab2   = denom1 + (size_t)N_NODES * 2;   // N x 256
    float* relW2  = xab2   + (size_t)N_NODES * 256; // NREL x 128
    float* num2   = relW2  + (size_t)NREL * 128;    // N x 128
    float* denom2 = num2   + (size_t)N_NODES * 128; // N

    float* out_rel = out + (size_t)N_NODES * 128;   // NREL x 128 output slot

    // zero accumulators (poisoned workspace; atomics require zero init)
    hipMemsetAsync(num1,   0, (size_t)N_NODES * 128 * sizeof(float), stream);
    hipMemsetAsync(denom1, 0, (size_t)N_NODES * 2   * sizeof(float), stream);
    hipMemsetAsync(num2,   0, (size_t)N_NODES * 128 * sizeof(float), stream);
    hipMemsetAsync(denom2, 0, (size_t)N_NODES       * sizeof(float), stream);

    // ---- Layer 1 projections (WMMA) ----
    // xnode[:, h*128 + part*64 .. +64] = x @ w1_{a,b}(h)^T
    for (int h = 0; h < 2; ++h)
        for (int part = 0; part < 2; ++part)
            launch_gemm<1>(stream, x, w1h + (size_t)h * 64 * 384 + part * 128,
                           xnode, N_NODES, 64, 128, 384, 256,
                           h * 128 + part * 64);
    // ec[:, h*64 ..] = edge_embed @ w1c(h)^T   (the one true edge GEMM)
    for (int h = 0; h < 2; ++h)
        launch_gemm<1>(stream, eemb, w1h + (size_t)h * 64 * 384 + 256,
                       ec, E_EDGES, 64, 128, 384, 128, h * 64);
    // relc[:, h*64 ..] = rel @ w1c(h)^T  (covers nhop edges)
    for (int h = 0; h < 2; ++h)
        launch_gemm<1>(stream, rel, w1h + (size_t)h * 64 * 384 + 256,
                       relc, NREL, 64, 128, 384, 128, h * 64);

    // ---- Layer 1 edge scatter + softmax-style aggregation ----
    {
        int blocks = (ET_EDGES + 7) / 8;
        gat_edge1<<<blocks, 256, 0, stream>>>(el, eln, etn, xnode, ec, relc,
                                              w2h, num1, denom1);
    }
    gat_fin1<<<(N_NODES * 128 + 255) / 256, 256, 0, stream>>>(num1, denom1);

    // ---- Layer 2 projections (WMMA) ----
    // out_relation = rel @ WR (B not transposed: LDBK=128, ldbn=1) -> d_out tail
    launch_gemm<128>(stream, rel, WR, out_rel, NREL, 128, 128, 1, 128, 0);
    // xab2[:, part*128 ..] = layer_x @ w1_{a2,b2}^T
    for (int part = 0; part < 2; ++part)
        launch_gemm<1>(stream, num1, w1o + part * 128,
                       xab2, N_NODES, 128, 128, 384, 256, part * 128);
    // relW2 = out_relation @ w1c2^T  (layer-2 edge GEMM collapses to table)
    launch_gemm<1>(stream, out_rel, w1o + 256, relW2, NREL, 128, 128, 384, 128, 0);

    // ---- Layer 2 edge scatter + finalize (double ELU) ----
    {
        int blocks = (ET_EDGES + 7) / 8;
        gat_edge2<<<blocks, 256, 0, stream>>>(el, eln, et, etn, xab2, relW2,
                                              w2o, num2, denom2);
    }
    gat_fin2<<<(N_NODES * 128 + 255) / 256, 256, 0, stream>>>(num2, denom2, out);
}